// ConditionalModuleBGR_50113678410581
// MI455X (gfx1250) — compile-verified
//
#include <hip/hip_runtime.h>
#include <hip/hip_bf16.h>

// ---------------------------------------------------------------------------
// ConditionalModuleBGR: per-channel 64-bin histogram ([16,3,1024,1024] fp32,
// 201 MB streamed -> memory bound, ~8.6us at 23.3 TB/s) followed by a tiny
// 192->128->32 MLP done with V_WMMA_F32_16X16X4_F32 (batch M=16 fits the
// 16x16 WMMA tile exactly).
// ---------------------------------------------------------------------------

typedef __attribute__((ext_vector_type(2))) float v2f;
typedef __attribute__((ext_vector_type(4))) float v4f;
typedef __attribute__((ext_vector_type(8))) float v8f;

#define N_IMG      16
#define N_CH       48          // 16 images * 3 channels
#define BINS       64
#define CH_ELEMS   (1024 * 1024)
#define BPC        64          // blocks per channel
#define HIST_WORDS (N_CH * BINS)   // 3072 u32

// ---------------------------------------------------------------------------
// Kernel 0: zero the histogram workspace (graph-safe, no hipMemset needed).
// ---------------------------------------------------------------------------
__global__ __launch_bounds__(256) void hist_zero_kernel(unsigned int* __restrict__ hist) {
    int i = blockIdx.x * blockDim.x + threadIdx.x;
    if (i < HIST_WORDS) hist[i] = 0u;
}

// ---------------------------------------------------------------------------
// Kernel 1: streaming histogram. One block handles 16384 elements of one
// (image,channel). 128-bit non-temporal loads (single-use stream), 4 LDS
// sub-histograms to spread bank/atomic pressure, one global atomic merge.
// ---------------------------------------------------------------------------
__global__ __launch_bounds__(256) void hist_kernel(const float* __restrict__ img,
                                                   unsigned int* __restrict__ hist) {
    __shared__ unsigned int sh[4][BINS];

    const int tid = threadIdx.x;
    // zero the 4*64 = 256 LDS counters (one per thread)
    ((unsigned int*)sh)[tid] = 0u;
    __syncthreads();

    const int ch    = blockIdx.x / BPC;
    const int chunk = blockIdx.x % BPC;

    const v4f* __restrict__ p =
        (const v4f*)(img + (size_t)ch * CH_ELEMS + (size_t)chunk * (CH_ELEMS / BPC));

    unsigned int* __restrict__ my = sh[tid & 3];

    // 16384 elems / (256 threads * 4) = 16 float4 per thread, coalesced.
#pragma unroll 4
    for (int i = 0; i < 16; ++i) {
        v4f v = __builtin_nontemporal_load(&p[i * 256 + tid]);
        int b0 = min(max((int)(v.x * 64.0f), 0), 63);
        int b1 = min(max((int)(v.y * 64.0f), 0), 63);
        int b2 = min(max((int)(v.z * 64.0f), 0), 63);
        int b3 = min(max((int)(v.w * 64.0f), 0), 63);
        atomicAdd(&my[b0], 1u);
        atomicAdd(&my[b1], 1u);
        atomicAdd(&my[b2], 1u);
        atomicAdd(&my[b3], 1u);
    }
    __syncthreads();

    if (tid < BINS) {
        unsigned int s = sh[0][tid] + sh[1][tid] + sh[2][tid] + sh[3][tid];
        if (s) atomicAdd(&hist[ch * BINS + tid], s);
    }
}

// ---------------------------------------------------------------------------
// Kernel 2: one-wave MLP using V_WMMA_F32_16X16X4_F32.
//
// hist workspace [48*64] u32 is exactly the row-major H[16][192] feature
// matrix (n*192 + c*64 + bin).
//
// WMMA f32 16x16x4 operand layouts (ISA 7.12.2):
//   A (16x4, MxK):  lane L<16:  v0 = A[M=L][K=0],  v1 = A[M=L][K=1]
//                   lane L>=16: v0 = A[M=L-16][K=2], v1 = A[M=L-16][K=3]
//   B (4x16, KxN):  lane L<16:  v0 = B[K=0][N=L],  v1 = B[K=1][N=L]
//                   lane L>=16: v0 = B[K=2][N=L-16], v1 = B[K=3][N=L-16]
//   C/D (16x16):    VGPR v, lane L<16 -> (M=v,   N=L)
//                           lane L>=16 -> (M=v+8, N=L-16)
// ---------------------------------------------------------------------------
__global__ __launch_bounds__(32) void mlp_wmma_kernel(const unsigned int* __restrict__ hist,
                                                      const float* __restrict__ params,
                                                      float* __restrict__ out) {
    __shared__ float H1[16][128];   // layer-1 activations

    const int lane = threadIdx.x;
    const int m    = lane & 15;     // row (A) / column (B,C,D) within half-wave
    const int kh   = lane >> 4;     // 0: K=0,1 half, 1: K=2,3 half

    const float* __restrict__ W1 = params;                 // [192][128]
    const float* __restrict__ b1 = params + 192 * 128;     // [128]
    const float* __restrict__ W2 = b1 + 128;               // [128][32]
    const float* __restrict__ b2 = W2 + 128 * 32;          // [32]
    const float  g               = b2[32];                 // global scalar param

    // ---- Layer 1: [16,192] x [192,128] + b1, ReLU -> LDS -----------------
    for (int n0 = 0; n0 < 128; n0 += 16) {
        v8f acc = {};
        for (int k0 = 0; k0 < 192; k0 += 4) {
            const int k = k0 + 2 * kh;
            v2f a, b;
            a.x = (float)hist[m * 192 + k];
            a.y = (float)hist[m * 192 + k + 1];
            b.x = W1[k * 128 + n0 + m];
            b.y = W1[(k + 1) * 128 + n0 + m];
            acc = __builtin_amdgcn_wmma_f32_16x16x4_f32(
                /*neg_a=*/false, a, /*neg_b=*/false, b,
                /*c_mod=*/(short)0, acc, /*reuse_a=*/false, /*reuse_b=*/false);
        }
        const float bias = b1[n0 + m];
#pragma unroll
        for (int v = 0; v < 8; ++v) {
            float x = acc[v] + bias;
            H1[v + 8 * kh][n0 + m] = x > 0.0f ? x : 0.0f;
        }
    }
    __syncthreads();   // single wave: cheap; orders LDS writes before reads

    // ---- Layer 2: [16,128] x [128,32] + b2, sigmoid(g + x) -> out --------
    for (int n0 = 0; n0 < 32; n0 += 16) {
        v8f acc = {};
        for (int k0 = 0; k0 < 128; k0 += 4) {
            const int k = k0 + 2 * kh;
            v2f a, b;
            a.x = H1[m][k];
            a.y = H1[m][k + 1];
            b.x = W2[k * 32 + n0 + m];
            b.y = W2[(k + 1) * 32 + n0 + m];
            acc = __builtin_amdgcn_wmma_f32_16x16x4_f32(
                false, a, false, b, (short)0, acc, false, false);
        }
        const float bias = b2[n0 + m];
#pragma unroll
        for (int v = 0; v < 8; ++v) {
            float x = g + acc[v] + bias;
            float s = 1.0f / (1.0f + __expf(-x));
            out[(v + 8 * kh) * 32 + n0 + m] = s;
        }
    }
}

// ---------------------------------------------------------------------------
extern "C" void kernel_launch(void* const* d_in, const int* in_sizes, int n_in,
                              void* d_out, int out_size, void* d_ws, size_t ws_size,
                              hipStream_t stream) {
    (void)in_sizes; (void)n_in; (void)out_size; (void)ws_size;

    const float* img    = (const float*)d_in[0];   // [16,3,1024,1024] f32
    const float* params = (const float*)d_in[1];   // [28833] f32
    float*       out    = (float*)d_out;           // [16,32] f32
    unsigned int* hist  = (unsigned int*)d_ws;     // [48*64] u32 scratch

    hist_zero_kernel<<<(HIST_WORDS + 255) / 256, 256, 0, stream>>>(hist);
    hist_kernel<<<N_CH * BPC, 256, 0, stream>>>(img, hist);
    mlp_wmma_kernel<<<1, 32, 0, stream>>>(hist, params, out);
}